// MedianPool2d_60533269070300
// MI455X (gfx1250) — compile-verified
//
#include <hip/hip_runtime.h>

// MedianPool2d 3x3, stride 1, zero-pad 1, fp32 (8,64,256,256).
// Memory-bound: 268 MB compulsory traffic @ 23.3 TB/s ~= 11.5 us floor.
// Strategy: async-stage an 18x258 input tile into LDS per block using
// CDNA5 GLOBAL_LOAD_ASYNC_TO_LDS_B128 (16B-aligned src+dst, 512 B per
// wave-instruction) + s_wait_asynccnt, then compute a 256x16 output strip
// with the separable median-of-9 decomposition (18 fp32 min/max per
// output after row-triple register reuse).

#define WIDTH  256   // H == W == 256
#define TH     16    // output rows per block
#define TROWS  (TH + 2)
#define LDSW   264   // padded LDS row stride (floats); 264*4 = 1056 B (16B mult.)
#define XOFF   4     // interior starts at LDS col 4 -> 16B-aligned b128 dests
#define NIMG   (8 * 64)

#if defined(__gfx1250__) && __has_builtin(__builtin_amdgcn_global_load_async_to_lds_b32)
#define HAVE_ASYNC_LDS 1
#else
#define HAVE_ASYNC_LDS 0
#endif

#if HAVE_ASYNC_LDS
typedef __attribute__((address_space(1))) int  gint_t;
typedef __attribute__((address_space(3))) int  lint_t;
#if __has_builtin(__builtin_amdgcn_global_load_async_to_lds_b128)
#define HAVE_ASYNC_B128 1
typedef int v4i_t __attribute__((ext_vector_type(4)));
typedef __attribute__((address_space(1))) v4i_t gv4i_t;
typedef __attribute__((address_space(3))) v4i_t lv4i_t;
#else
#define HAVE_ASYNC_B128 0
#endif
#endif

__device__ __forceinline__ void sort3f(float a, float b, float c,
                                       float& lo, float& mid, float& hi) {
    float t1 = fminf(a, b);
    float t2 = fmaxf(a, b);
    lo  = fminf(t1, c);
    hi  = fmaxf(t2, c);
    mid = fmaxf(t1, fminf(t2, c));
}

__device__ __forceinline__ float med3f(float a, float b, float c) {
    return fmaxf(fminf(a, b), fminf(fmaxf(a, b), c));
}

__global__ __launch_bounds__(256)
void median3x3_kernel(const float* __restrict__ in, float* __restrict__ out) {
    __shared__ float tile[TROWS * LDSW];

    const int tid   = threadIdx.x;   // output column x = tid, 0..255
    const int strip = blockIdx.x;    // 0..15
    const int bc    = blockIdx.y;    // 0..511 (B*C flattened)
    const int y0    = strip * TH;

    const float* __restrict__ img  = in  + (size_t)bc * WIDTH * WIDTH;
    float*       __restrict__ outp = out + (size_t)bc * WIDTH * WIDTH;

    // ---- Phase A: zero the halo cells the async loads won't cover ----
    // left/right halo columns (LDS cols XOFF-1 and XOFF+WIDTH) for every row
    for (int r = tid; r < TROWS; r += 256) {
        tile[r * LDSW + XOFF - 1]     = 0.0f;
        tile[r * LDSW + XOFF + WIDTH] = 0.0f;
    }
    // top / bottom padding rows only at the image boundary strips
    if (y0 == 0) {
        for (int c = tid; c < WIDTH + 2; c += 256)
            tile[XOFF - 1 + c] = 0.0f;
    }
    if (y0 + TH == WIDTH) {
        for (int c = tid; c < WIDTH + 2; c += 256)
            tile[(TROWS - 1) * LDSW + XOFF - 1 + c] = 0.0f;
    }

    // ---- Phase B: stage 18 input rows (interior columns) into LDS ----
#if HAVE_ASYNC_LDS && HAVE_ASYNC_B128
    // b128 path: 64 lanes cover one 1 KB row; 256 threads stage 4 rows per
    // pass. All predicates depend only on tid/64 (constant per wave32) and
    // block-uniform bounds -> EXEC all-ones for every issued async load.
    {
        const int lane4 = tid & 63;   // 16B chunk within the row
        const int rsub  = tid >> 6;   // 0..3, uniform within a wave
#pragma unroll
        for (int rb = 0; rb < 20; rb += 4) {
            const int r  = rb + rsub;
            const int gr = y0 - 1 + r;
            if (r < TROWS && gr >= 0 && gr < WIDTH) {
                const float* gsrc = img + (size_t)gr * WIDTH + lane4 * 4;
                float*       ldst = &tile[r * LDSW + XOFF + lane4 * 4];
                __builtin_amdgcn_global_load_async_to_lds_b128(
                    (gv4i_t*)gsrc, (lv4i_t*)ldst, /*offset=*/0, /*cpol=*/0);
            }
        }
    }
#else
#pragma unroll
    for (int r = 0; r < TROWS; ++r) {
        const int gr = y0 - 1 + r;
        if (gr >= 0 && gr < WIDTH) {
            const float* gsrc = img + (size_t)gr * WIDTH + tid;
            float*       ldst = &tile[r * LDSW + XOFF + tid];
#if HAVE_ASYNC_LDS
            __builtin_amdgcn_global_load_async_to_lds_b32(
                (gint_t*)gsrc, (lint_t*)ldst, /*offset=*/0, /*cpol=*/0);
#else
            *ldst = __builtin_nontemporal_load(gsrc);
#endif
        }
    }
#endif

#if HAVE_ASYNC_LDS
#if __has_builtin(__builtin_amdgcn_s_wait_asynccnt)
    __builtin_amdgcn_s_wait_asynccnt(0);
#else
    asm volatile("s_wait_asynccnt 0" ::: "memory");
#endif
#endif
    __syncthreads();

    // ---- Phase C: separable median-of-9 over the LDS tile ----
    // Row triple (x-1, x, x+1) sorted once per tile row, reused by the
    // 3 vertical windows through register rotation.
    float lo0, mid0, hi0, lo1, mid1, hi1, lo2, mid2, hi2;
    {
        float a = tile[0 * LDSW + tid + XOFF - 1];
        float b = tile[0 * LDSW + tid + XOFF + 0];
        float c = tile[0 * LDSW + tid + XOFF + 1];
        sort3f(a, b, c, lo0, mid0, hi0);
        a = tile[1 * LDSW + tid + XOFF - 1];
        b = tile[1 * LDSW + tid + XOFF + 0];
        c = tile[1 * LDSW + tid + XOFF + 1];
        sort3f(a, b, c, lo1, mid1, hi1);
    }

#pragma unroll
    for (int r = 0; r < TH; ++r) {
        float a = tile[(r + 2) * LDSW + tid + XOFF - 1];
        float b = tile[(r + 2) * LDSW + tid + XOFF + 0];
        float c = tile[(r + 2) * LDSW + tid + XOFF + 1];
        sort3f(a, b, c, lo2, mid2, hi2);

        float maxlo = fmaxf(fmaxf(lo0, lo1), lo2);
        float minhi = fminf(fminf(hi0, hi1), hi2);
        float medmi = med3f(mid0, mid1, mid2);
        float m     = med3f(maxlo, medmi, minhi);

        outp[(size_t)(y0 + r) * WIDTH + tid] = m;

        lo0 = lo1; mid0 = mid1; hi0 = hi1;
        lo1 = lo2; mid1 = mid2; hi1 = hi2;
    }
}

extern "C" void kernel_launch(void* const* d_in, const int* in_sizes, int n_in,
                              void* d_out, int out_size, void* d_ws, size_t ws_size,
                              hipStream_t stream) {
    (void)in_sizes; (void)n_in; (void)out_size; (void)d_ws; (void)ws_size;
    const float* x = (const float*)d_in[0];
    float*       y = (float*)d_out;

    dim3 grid(WIDTH / TH, NIMG);   // (16 strips, 512 images)
    dim3 block(256);
    median3x3_kernel<<<grid, block, 0, stream>>>(x, y);
}